// RTDETRPostProcessor_75350906241476
// MI455X (gfx1250) — compile-verified
//
#include <hip/hip_runtime.h>
#include <hip/hip_bf16.h>
#include <stdint.h>

#define BATCH    256
#define QN       1000
#define CN       80
#define QC       80000          // QN*CN
#define TOPK     300
#define NBINS    2048
#define CHUNK    4000           // floats per staged chunk
#define NCHUNK   20             // 20*4000 = 80000
#define NLOADERS 1000           // threads staging 4 floats (16B) each
#define NTHREADS 1024
#define SORTN    512
#define CANDCAP  1024           // threshold-bin candidate capacity

__device__ __forceinline__ uint32_t orderKey(float x) {
  uint32_t u = __float_as_uint(x);
  return (u & 0x80000000u) ? ~u : (u | 0x80000000u);
}
__device__ __forceinline__ float keyToFloat(uint32_t u) {
  uint32_t v = (u & 0x80000000u) ? (u & 0x7fffffffu) : ~u;
  return __uint_as_float(v);
}

// In-place suffix scan over 2048 bins: h[i] = sum_{j>=i} h[j]. All threads call.
__device__ __forceinline__ void suffixScan2048(uint32_t* h, int t) {
  for (int off = 1; off < NBINS; off <<= 1) {
    const int i0 = t, i1 = t + NTHREADS;
    uint32_t a0 = (i0 + off < NBINS) ? h[i0 + off] : 0u;
    uint32_t a1 = (i1 + off < NBINS) ? h[i1 + off] : 0u;
    __syncthreads();
    h[i0] += a0;
    h[i1] += a1;
    __syncthreads();
  }
}

__global__ __launch_bounds__(NTHREADS)
void rtdetr_post_kernel(const float* __restrict__ logits,
                        const float* __restrict__ boxes,
                        const float* __restrict__ sizes,
                        float* __restrict__ out) {
  __shared__ __align__(16) float lbuf[2][CHUNK];     // 32000 B staging
  __shared__ uint32_t hist[2][NBINS];                // 16384 B (wave-privatized)
  __shared__ uint64_t cand[CANDCAP];                 // 8192 B threshold-bin pool
  __shared__ uint64_t sortp[SORTN];                  // 4096 B packed (key|~idx)
  __shared__ int cntA, cntC, cntB1, cntB2;
  __shared__ int threshBin, nAboveSh, thresh2, nAbove2;

  const int t = threadIdx.x;
  const int b = blockIdx.x;
  const float* blogits = logits + (size_t)b * QC;

  for (int i = t; i < 2 * NBINS; i += NTHREADS) ((uint32_t*)hist)[i] = 0u;
  if (t < SORTN) sortp[t] = 0ull;
  if (t == 0) { cntA = 0; cntC = 0; cntB1 = 0; cntB2 = 0; }
  __syncthreads();

  const uint32_t ldsBase = (uint32_t)(uintptr_t)&lbuf[0][0];
  const int hcopy = (t >> 5) & 1;                    // wave-parity privatization

  // ---------------- Pass 1: histogram (async-LDS double-buffered stream) ---
  if (t < NLOADERS) {
    const uint64_t g0 = (uint64_t)(uintptr_t)(blogits + t * 4);
    const uint32_t l0 = ldsBase + (uint32_t)(t * 16);
    const uint32_t l1 = l0 + CHUNK * 4;
    asm volatile("global_load_async_to_lds_b128 %0, %1, off"
                 :: "v"(l0), "v"(g0) : "memory");
    for (int ch = 0; ch < NCHUNK; ++ch) {
      if (ch + 1 < NCHUNK) {
        uint64_t g = g0 + (uint64_t)(ch + 1) * (CHUNK * 4);
        uint32_t l = ((ch + 1) & 1) ? l1 : l0;
        asm volatile("global_load_async_to_lds_b128 %0, %1, off"
                     :: "v"(l), "v"(g) : "memory");
        asm volatile("s_wait_asynccnt 1" ::: "memory");
      } else {
        asm volatile("s_wait_asynccnt 0" ::: "memory");
      }
      const float* p = &lbuf[ch & 1][t * 4];
#pragma unroll
      for (int j = 0; j < 4; ++j) {
        uint32_t key = orderKey(p[j]);
        atomicAdd(&hist[hcopy][key >> 21], 1u);
      }
    }
  }
  __syncthreads();

  // merge privatized copies, then suffix scan copy 0
  for (int i = t; i < NBINS; i += NTHREADS) hist[0][i] += hist[1][i];
  __syncthreads();
  suffixScan2048(hist[0], t);
  for (int i = t; i < NBINS; i += NTHREADS) {
    uint32_t ge = hist[0][i];
    uint32_t gt = (i + 1 < NBINS) ? hist[0][i + 1] : 0u;
    if (ge >= (uint32_t)TOPK && gt < (uint32_t)TOPK) {
      threshBin = i; nAboveSh = (int)gt;
    }
  }
  __syncthreads();
  const int T  = threshBin;
  const int nA = nAboveSh;

  // ---------------- Pass 2: collect (async-LDS stream, L2-resident) --------
  if (t < NLOADERS) {
    const uint64_t g0 = (uint64_t)(uintptr_t)(blogits + t * 4);
    const uint32_t l0 = ldsBase + (uint32_t)(t * 16);
    const uint32_t l1 = l0 + CHUNK * 4;
    asm volatile("global_load_async_to_lds_b128 %0, %1, off"
                 :: "v"(l0), "v"(g0) : "memory");
    for (int ch = 0; ch < NCHUNK; ++ch) {
      if (ch + 1 < NCHUNK) {
        uint64_t g = g0 + (uint64_t)(ch + 1) * (CHUNK * 4);
        uint32_t l = ((ch + 1) & 1) ? l1 : l0;
        asm volatile("global_load_async_to_lds_b128 %0, %1, off"
                     :: "v"(l), "v"(g) : "memory");
        asm volatile("s_wait_asynccnt 1" ::: "memory");
      } else {
        asm volatile("s_wait_asynccnt 0" ::: "memory");
      }
      const float* p = &lbuf[ch & 1][t * 4];
      const uint32_t base = (uint32_t)(ch * CHUNK + t * 4);
#pragma unroll
      for (int j = 0; j < 4; ++j) {
        uint32_t key = orderKey(p[j]);
        int bin = (int)(key >> 21);
        uint64_t packed = ((uint64_t)key << 32) | (uint64_t)(0xFFFFFFFFu - (base + j));
        if (bin > T) {
          int pos = atomicAdd(&cntA, 1);           // exactly nA of these, nA < K
          sortp[pos] = packed;
        } else if (bin == T) {
          int c = atomicAdd(&cntC, 1);
          if (c < CANDCAP) cand[c] = packed;
        }
      }
    }
  }
  __syncthreads();

  // ---------------- Refinement: radix level 2 inside threshold bin ---------
  for (int i = t; i < NBINS; i += NTHREADS) hist[0][i] = 0u;
  __syncthreads();
  const int nc = (cntC < CANDCAP) ? cntC : CANDCAP;
  for (int i = t; i < nc; i += NTHREADS) {
    uint32_t key = (uint32_t)(cand[i] >> 32);
    atomicAdd(&hist[0][(key >> 10) & 0x7FFu], 1u);  // key bits [20:10]
  }
  __syncthreads();
  suffixScan2048(hist[0], t);
  const int K2 = TOPK - nA;                          // 1 <= K2 <= TOPK
  for (int i = t; i < NBINS; i += NTHREADS) {
    uint32_t ge = hist[0][i];
    uint32_t gt = (i + 1 < NBINS) ? hist[0][i + 1] : 0u;
    if (ge >= (uint32_t)K2 && gt < (uint32_t)K2) {
      thresh2 = i; nAbove2 = (int)gt;
    }
  }
  __syncthreads();
  const int T2 = thresh2, nA2 = nAbove2;
  for (int i = t; i < nc; i += NTHREADS) {
    uint64_t p = cand[i];
    int sub = (int)(((uint32_t)(p >> 32) >> 10) & 0x7FFu);
    if (sub > T2) {
      int pos = nA + atomicAdd(&cntB1, 1);           // exactly nA2 of these
      sortp[pos] = p;
    } else if (sub == T2) {
      int pos = nA + nA2 + atomicAdd(&cntB2, 1);
      if (pos < TOPK) sortp[pos] = p;
    }
  }
  __syncthreads();

  // ---------------- Bitonic sort 512 packed entries, descending ------------
  for (int k = 2; k <= SORTN; k <<= 1) {
    for (int j = k >> 1; j > 0; j >>= 1) {
      if (t < SORTN) {
        int i = t, ixj = i ^ j;
        if (ixj > i) {
          bool up = ((i & k) == 0);
          uint64_t pi = sortp[i], pj = sortp[ixj];
          if ((pi < pj) == up) {                     // descending comparator
            sortp[i] = pj; sortp[ixj] = pi;
          }
        }
      }
      __syncthreads();
    }
  }

  // ---------------- Decode + write outputs ---------------------------------
  if (t < TOPK) {
    uint64_t p   = sortp[t];
    uint32_t key = (uint32_t)(p >> 32);
    uint32_t idx = 0xFFFFFFFFu - (uint32_t)p;
    float logit = keyToFloat(key);
    float score = 1.0f / (1.0f + expf(-logit));
    uint32_t label = idx % CN;
    uint32_t q     = idx / CN;
    const float4 bx = *(const float4*)(boxes + ((size_t)b * QN + q) * 4);
    float W = sizes[b * 2 + 0];
    float H = sizes[b * 2 + 1];
    float hw = 0.5f * bx.z, hh = 0.5f * bx.w;
    int o = b * TOPK + t;
    // layout: labels [B*K] | boxes [B*K*4] | scores [B*K]
    out[o] = (float)label;
    float* ob = out + BATCH * TOPK + (size_t)o * 4;
    ob[0] = (bx.x - hw) * W;
    ob[1] = (bx.y - hh) * H;
    ob[2] = (bx.x + hw) * W;
    ob[3] = (bx.y + hh) * H;
    out[BATCH * TOPK + BATCH * TOPK * 4 + o] = score;
  }
}

extern "C" void kernel_launch(void* const* d_in, const int* in_sizes, int n_in,
                              void* d_out, int out_size, void* d_ws, size_t ws_size,
                              hipStream_t stream) {
  const float* logits = (const float*)d_in[0];   // [256,1000,80]
  const float* boxes  = (const float*)d_in[1];   // [256,1000,4]
  const float* sizes  = (const float*)d_in[2];   // [256,2]
  float* out = (float*)d_out;                    // labels|boxes|scores flat
  (void)in_sizes; (void)n_in; (void)out_size; (void)d_ws; (void)ws_size;
  rtdetr_post_kernel<<<BATCH, NTHREADS, 0, stream>>>(logits, boxes, sizes, out);
}